// GCNConvNetBN_62981400428924
// MI455X (gfx1250) — compile-verified
//
#include <hip/hip_runtime.h>

typedef __attribute__((ext_vector_type(2))) float v2f;
typedef __attribute__((ext_vector_type(8))) float v8f;

#define BN_EPS 1e-5f

// ---------------- utility ----------------
__global__ void k_fill(float* __restrict__ p, int n, float v) {
    int t = blockIdx.x * blockDim.x + threadIdx.x;
    if (t < n) p[t] = v;
}

__global__ void k_degree(const int* __restrict__ dst, float* __restrict__ deg, int E) {
    int e = blockIdx.x * blockDim.x + threadIdx.x;
    if (e < E) atomicAdd(deg + dst[e], 1.0f);
}

__global__ void k_dinv(const float* __restrict__ deg, float* __restrict__ dinv, int N) {
    int n = blockIdx.x * blockDim.x + threadIdx.x;
    if (n < N) dinv[n] = rsqrtf(deg[n]);   // deg >= 2 always (two self loops)
}

// pout[n][c] = 2*dinv[n]^2 * h[n][c]   (two self-loops, each norm = dinv*dinv)
__global__ void k_selfloop(const float* __restrict__ Hin, const float* __restrict__ scale,
                           const float* __restrict__ shift, const float* __restrict__ dinv,
                           float* __restrict__ Pout, int N, int C, int affine) {
    int t = blockIdx.x * blockDim.x + threadIdx.x;
    if (t >= N * C) return;
    int n = t / C, c = t - n * C;
    float h = Hin[t];
    if (affine) h = h * scale[c] + shift[c];
    float d = dinv[n];
    Pout[t] = 2.0f * d * d * h;
}

// pout[dst][c] += dinv[src]*dinv[dst] * (h[src][c]*scale+shift)
__global__ void k_edge_prop(const int* __restrict__ esrc, const int* __restrict__ edst,
                            const float* __restrict__ dinv, const float* __restrict__ Hin,
                            const float* __restrict__ scale, const float* __restrict__ shift,
                            float* __restrict__ Pout, int E, int C, int affine) {
    if (C >= 32) {  // one wave per edge, lanes stride channels
        int wave = (blockIdx.x * blockDim.x + threadIdx.x) >> 5;
        int lane = threadIdx.x & 31;
        if (wave >= E) return;
        int s = esrc[wave], d = edst[wave];
        float w = dinv[s] * dinv[d];
        const float* hs = Hin + (size_t)s * C;
        float* pd = Pout + (size_t)d * C;
        for (int c = lane; c < C; c += 32) {
            float h = hs[c];
            if (affine) h = h * scale[c] + shift[c];
            atomicAdd(pd + c, w * h);
        }
    } else {        // tiny C (first layer, C=3): one thread per edge
        int e = blockIdx.x * blockDim.x + threadIdx.x;
        if (e >= E) return;
        int s = esrc[e], d = edst[e];
        float w = dinv[s] * dinv[d];
        const float* hs = Hin + (size_t)s * C;
        float* pd = Pout + (size_t)d * C;
        for (int c = 0; c < C; c++) {
            float h = hs[c];
            if (affine) h = h * scale[c] + shift[c];
            atomicAdd(pd + c, w * h);
        }
    }
}

// layer-1 GEMM, K=3, C=64: vector kernel with fused bias+relu+BN stats
__global__ __launch_bounds__(256) void k_gemm_k3(const float* __restrict__ P,
                                                 const float* __restrict__ W,
                                                 const float* __restrict__ bias,
                                                 float* __restrict__ Hout,
                                                 float* __restrict__ colsum,
                                                 float* __restrict__ colsq, int N) {
    __shared__ float sS[256];
    __shared__ float sQ[256];
    int c = threadIdx.x & 63;
    int rl = threadIdx.x >> 6;
    float w0 = W[c], w1 = W[64 + c], w2 = W[128 + c];
    float b = bias[c];
    float s = 0.f, q = 0.f;
    int base = blockIdx.x * 256;
    for (int r = rl; r < 256; r += 4) {
        int n = base + r;
        if (n < N) {
            const float* p = P + (size_t)n * 3;
            float v = fmaf(p[0], w0, fmaf(p[1], w1, fmaf(p[2], w2, b)));
            v = v > 0.f ? v : 0.f;
            Hout[(size_t)n * 64 + c] = v;
            s += v; q += v * v;
        }
    }
    sS[threadIdx.x] = s; sQ[threadIdx.x] = q;
    __syncthreads();
    if (rl == 0) {
        s = sS[c] + sS[64 + c] + sS[128 + c] + sS[192 + c];
        q = sQ[c] + sQ[64 + c] + sQ[128 + c] + sQ[192 + c];
        atomicAdd(colsum + c, s);
        atomicAdd(colsq + c, q);
    }
}

// WMMA fp32 GEMM: H = relu(P[N,K] @ W[K,C] + bias), with fused BN-stat
// reduction and (optionally) fused per-graph max/min pooling of pre-BN values.
// Block = 8 waves = 2(M) x 4(N) tiles of 16x16 -> block tile 32 x 64.
__global__ __launch_bounds__(256) void k_wmma_gemm(
    const float* __restrict__ P, const float* __restrict__ W,
    const float* __restrict__ bias, float* __restrict__ Hout,
    float* __restrict__ colsum, float* __restrict__ colsq,
    float* __restrict__ gmax, float* __restrict__ gmin,
    const int* __restrict__ batch, int K, int C) {
    const int lane = threadIdx.x & 31;
    const int wave = threadIdx.x >> 5;
    const int wm = wave & 1, wn = wave >> 1;
    const int m0 = blockIdx.x * 32 + wm * 16;
    const int n0 = blockIdx.y * 64 + wn * 16;
    const bool hilane = lane >= 16;
    const int l = lane & 15;
    const int koff = hilane ? 2 : 0;
    // A (16x4 f32): lanes 0-15 hold K=k+0,k+1 of row m0+l; lanes 16-31 K=k+2,k+3
    const float* Ap = P + (size_t)(m0 + l) * K + koff;
    // B (4x16 f32): lane holds column n0+l; low lanes K=k+0,k+1, high lanes K=k+2,k+3
    const float* Bp = W + (size_t)koff * C + (n0 + l);

    v8f acc = {};
    for (int k = 0; k < K; k += 4) {
        v2f a, b;
        a[0] = Ap[k];
        a[1] = Ap[k + 1];
        b[0] = Bp[(size_t)k * C];
        b[1] = Bp[(size_t)(k + 1) * C];
        acc = __builtin_amdgcn_wmma_f32_16x16x4_f32(false, a, false, b, (short)0, acc,
                                                    false, false);
    }

    // C/D layout: VGPR v -> row m0 + v + (hilane ? 8 : 0), column n0+l.
    const int col = n0 + l;
    const float bb = bias[col];
    const int rbase = m0 + (hilane ? 8 : 0);
    float vals[8];
    float s = 0.f, q = 0.f, mx = -__FLT_MAX__, mn = __FLT_MAX__;
#pragma unroll
    for (int v = 0; v < 8; v++) {
        float r = acc[v] + bb;
        r = r > 0.f ? r : 0.f;
        vals[v] = r;
        s += r; q += r * r;
        mx = fmaxf(mx, r); mn = fminf(mn, r);
    }

    if (Hout) {
#pragma unroll
        for (int v = 0; v < 8; v++) Hout[(size_t)(rbase + v) * C + col] = vals[v];
    }

    // BN stats: combine the two lane-halves sharing a column, one atomic per column/wave
    s += __shfl_xor(s, 16, 32);
    q += __shfl_xor(q, 16, 32);
    if (lane < 16) {
        atomicAdd(colsum + col, s);
        atomicAdd(colsq + col, q);
    }

    if (gmax) {  // fused segment max/min of pre-BN relu values (>=0 -> uint pun is order-safe)
        int g0 = batch[m0], g1 = batch[m0 + 15];
        if (g0 == g1) {  // whole 16-row tile in one graph (common: ~1562 nodes/graph)
            float wmx = fmaxf(mx, __shfl_xor(mx, 16, 32));
            float wmn = fminf(mn, __shfl_xor(mn, 16, 32));
            if (lane < 16) {
                atomicMax((unsigned int*)(gmax + (size_t)g0 * C + col), __float_as_uint(wmx));
                atomicMin((unsigned int*)(gmin + (size_t)g0 * C + col), __float_as_uint(wmn));
            }
        } else {         // graph boundary inside tile: per-row
#pragma unroll
            for (int v = 0; v < 8; v++) {
                int gid = batch[rbase + v];
                atomicMax((unsigned int*)(gmax + (size_t)gid * C + col), __float_as_uint(vals[v]));
                atomicMin((unsigned int*)(gmin + (size_t)gid * C + col), __float_as_uint(vals[v]));
            }
        }
    }
}

// mean/var -> fold batchnorm into per-channel scale/shift
__global__ void k_bn_finalize(const float* __restrict__ colsum, const float* __restrict__ colsq,
                              const float* __restrict__ g, const float* __restrict__ be,
                              float* __restrict__ scale, float* __restrict__ shift,
                              int C, float invN) {
    int c = blockIdx.x * blockDim.x + threadIdx.x;
    if (c >= C) return;
    float m = colsum[c] * invN;
    float var = colsq[c] * invN - m * m;
    float sc = g[c] * rsqrtf(var + BN_EPS);
    scale[c] = sc;
    shift[c] = be[c] - m * sc;
}

// pooled[g][c] = BN(max over nodes) = (scale>=0 ? max : min)*scale + shift
__global__ void k_pool_combine(const float* __restrict__ gmax, const float* __restrict__ gmin,
                               const float* __restrict__ scale, const float* __restrict__ shift,
                               float* __restrict__ out, int total, int C) {
    int t = blockIdx.x * blockDim.x + threadIdx.x;
    if (t >= total) return;
    int c = t % C;
    float sc = scale[c];
    float v = sc >= 0.f ? gmax[t] : gmin[t];
    out[t] = v * sc + shift[c];
}

// tiny MLP GEMM (G=64 rows): relu(X@W + b)
__global__ void k_mlp_gemm(const float* __restrict__ X, const float* __restrict__ W,
                           const float* __restrict__ bias, float* __restrict__ Y,
                           int G, int K, int C) {
    int t = blockIdx.x * blockDim.x + threadIdx.x;
    if (t >= G * C) return;
    int r = t / C, c = t - r * C;
    float acc = bias[c];
    const float* x = X + (size_t)r * K;
    for (int k = 0; k < K; k++) acc = fmaf(x[k], W[(size_t)k * C + c], acc);
    Y[t] = acc > 0.f ? acc : 0.f;
}

// in-place BatchNorm over G=64 rows, one thread per channel
__global__ void k_mlp_bn(float* __restrict__ X, const float* __restrict__ g,
                         const float* __restrict__ be, int G, int C) {
    int c = blockIdx.x * blockDim.x + threadIdx.x;
    if (c >= C) return;
    float m = 0.f;
    for (int r = 0; r < G; r++) m += X[(size_t)r * C + c];
    m /= (float)G;
    float v = 0.f;
    for (int r = 0; r < G; r++) { float d = X[(size_t)r * C + c] - m; v += d * d; }
    v /= (float)G;
    float sc = g[c] * rsqrtf(v + BN_EPS);
    float sh = be[c] - m * sc;
    for (int r = 0; r < G; r++) X[(size_t)r * C + c] = X[(size_t)r * C + c] * sc + sh;
}

__global__ void k_fc(const float* __restrict__ X, const float* __restrict__ W,
                     const float* __restrict__ b, float* __restrict__ Y,
                     int G, int K, int C) {
    int t = blockIdx.x * blockDim.x + threadIdx.x;
    if (t >= G * C) return;
    int r = t / C, c = t - r * C;
    float acc = b[c];
    for (int k = 0; k < K; k++) acc = fmaf(X[(size_t)r * K + k], W[(size_t)k * C + c], acc);
    Y[t] = acc;
}

// ---------------- host orchestration ----------------
extern "C" void kernel_launch(void* const* d_in, const int* in_sizes, int n_in,
                              void* d_out, int out_size, void* d_ws, size_t ws_size,
                              hipStream_t stream) {
    (void)n_in; (void)out_size; (void)ws_size;
    const int N = in_sizes[0] / 3;       // 100000
    const int E = in_sizes[1] / 2;       // 1600000
    const int G = 64;

    const float* x    = (const float*)d_in[0];
    const int* esrc   = (const int*)d_in[1];
    const int* edst   = esrc + E;
    const int* batch  = (const int*)d_in[2];
    int idx = 3;
    const float *cw[5], *cb[5], *cg[5], *cbe[5];
    for (int i = 0; i < 5; i++) {
        cw[i]  = (const float*)d_in[idx++];
        cb[i]  = (const float*)d_in[idx++];
        cg[i]  = (const float*)d_in[idx++];
        cbe[i] = (const float*)d_in[idx++];
    }
    const float *mw[3], *mb[3], *mg[3], *mbe[3];
    for (int i = 0; i < 3; i++) {
        mw[i]  = (const float*)d_in[idx++];
        mb[i]  = (const float*)d_in[idx++];
        mg[i]  = (const float*)d_in[idx++];
        mbe[i] = (const float*)d_in[idx++];
    }
    const float* fcw = (const float*)d_in[idx++];
    const float* fcb = (const float*)d_in[idx++];
    float* out = (float*)d_out;

    // workspace carve-out (~130 MB; fits mostly in the 192 MB L2)
    size_t off = 0;
    auto carve = [&](size_t bytes) {
        void* p = (char*)d_ws + off;
        off += (bytes + 255) & ~(size_t)255;
        return p;
    };
    float* deg    = (float*)carve((size_t)N * 4);
    float* dinv   = (float*)carve((size_t)N * 4);
    float* pout   = (float*)carve((size_t)N * 128 * 4);
    float* hA     = (float*)carve((size_t)N * 64 * 4);     // layers with 64 out-ch
    float* hB     = (float*)carve((size_t)N * 128 * 4);    // layers with <=128 out-ch
    float* colsum = (float*)carve(1024 * 4);
    float* colsq  = (float*)carve(1024 * 4);
    float* scaleB = (float*)carve(1024 * 4);
    float* shiftB = (float*)carve(1024 * 4);
    float* gmax   = (float*)carve((size_t)G * 1024 * 4);
    float* gmin   = (float*)carve((size_t)G * 1024 * 4);
    float* mbuf0  = (float*)carve((size_t)G * 1024 * 4);
    float* mbuf1  = (float*)carve((size_t)G * 512 * 4);
    float* mbuf2  = (float*)carve((size_t)G * 256 * 4);
    float* mbuf3  = (float*)carve((size_t)G * 40 * 4);

    auto cdiv = [](int a, int b) { return (a + b - 1) / b; };

    // degrees (2 self loops per node) and dinv
    k_fill<<<cdiv(N, 256), 256, 0, stream>>>(deg, N, 2.0f);
    k_degree<<<cdiv(E, 256), 256, 0, stream>>>(edst, deg, E);
    k_dinv<<<cdiv(N, 256), 256, 0, stream>>>(deg, dinv, N);

    const int convC[6] = {3, 64, 64, 64, 128, 1024};
    float* hdst[5] = {hA, hB, hA, hB, nullptr};  // layer 4 output is fused into pooling
    const float* hin = x;

    for (int layer = 0; layer < 5; layer++) {
        int Ci = convC[layer], Co = convC[layer + 1];
        int affine = (layer > 0) ? 1 : 0;  // apply previous layer's BN during gather

        // propagate: pout = A_hat @ BN(hin)
        k_selfloop<<<cdiv(N * Ci, 256), 256, 0, stream>>>(hin, scaleB, shiftB, dinv,
                                                          pout, N, Ci, affine);
        if (Ci >= 32)
            k_edge_prop<<<cdiv(E * 32, 256), 256, 0, stream>>>(esrc, edst, dinv, hin,
                                                               scaleB, shiftB, pout, E, Ci, affine);
        else
            k_edge_prop<<<cdiv(E, 256), 256, 0, stream>>>(esrc, edst, dinv, hin,
                                                          scaleB, shiftB, pout, E, Ci, affine);

        // transform: relu(pout @ W + b), fused BN stats (and pooling on last layer)
        k_fill<<<cdiv(Co, 256), 256, 0, stream>>>(colsum, Co, 0.f);
        k_fill<<<cdiv(Co, 256), 256, 0, stream>>>(colsq, Co, 0.f);
        if (layer == 0) {
            k_gemm_k3<<<cdiv(N, 256), 256, 0, stream>>>(pout, cw[0], cb[0], hA,
                                                        colsum, colsq, N);
        } else if (layer < 4) {
            k_wmma_gemm<<<dim3(N / 32, Co / 64), 256, 0, stream>>>(
                pout, cw[layer], cb[layer], hdst[layer], colsum, colsq,
                nullptr, nullptr, batch, Ci, Co);
        } else {  // layer 4: 128 -> 1024, fused per-graph max/min pooling
            k_fill<<<cdiv(G * 1024, 256), 256, 0, stream>>>(gmax, G * 1024, 0.0f);
            k_fill<<<cdiv(G * 1024, 256), 256, 0, stream>>>(gmin, G * 1024, __FLT_MAX__);
            k_wmma_gemm<<<dim3(N / 32, 1024 / 64), 256, 0, stream>>>(
                pout, cw[4], cb[4], nullptr, colsum, colsq, gmax, gmin, batch, Ci, Co);
        }
        k_bn_finalize<<<cdiv(Co, 256), 256, 0, stream>>>(colsum, colsq, cg[layer], cbe[layer],
                                                         scaleB, shiftB, Co, 1.0f / (float)N);
        hin = hdst[layer];
    }

    // pooled = BN(segment_max)  ->  [64, 1024]
    k_pool_combine<<<cdiv(G * 1024, 256), 256, 0, stream>>>(gmax, gmin, scaleB, shiftB,
                                                            mbuf0, G * 1024, 1024);

    // MLP: 1024->512->256->40, each BN(relu(x@w+b))
    const int mlpC[4] = {1024, 512, 256, 40};
    float* mbufs[4] = {mbuf0, mbuf1, mbuf2, mbuf3};
    for (int i = 0; i < 3; i++) {
        k_mlp_gemm<<<cdiv(G * mlpC[i + 1], 256), 256, 0, stream>>>(
            mbufs[i], mw[i], mb[i], mbufs[i + 1], G, mlpC[i], mlpC[i + 1]);
        k_mlp_bn<<<cdiv(mlpC[i + 1], 256), 256, 0, stream>>>(mbufs[i + 1], mg[i], mbe[i],
                                                             G, mlpC[i + 1]);
    }

    // final fc: [64,40] @ [40,2] + b
    k_fc<<<1, 128, 0, stream>>>(mbuf3, fcw, fcb, out, G, 40, 2);
}